// SparseNet3D_70729521431108
// MI455X (gfx1250) — compile-verified
//
#include <hip/hip_runtime.h>

typedef float v2f __attribute__((ext_vector_type(2)));
typedef float v8f __attribute__((ext_vector_type(8)));

#define SLOPE 0.01f
#define BN_EPS 1e-5f

// ---------------------------------------------------------------------------
// Stage 0: binarize mask, zero-fill inactive voxels of feat (C=4 fixed).
// ---------------------------------------------------------------------------
__global__ void k_binarize(const float* __restrict__ feat,
                           const float* __restrict__ mask,
                           float* __restrict__ x0, float* __restrict__ m0,
                           int S) {
  int i = blockIdx.x * 256 + threadIdx.x;   // over B*S, B=2
  if (i >= 2 * S) return;
  int b = i / S, s = i - b * S;
  float mv = mask[i] > 0.5f ? 1.0f : 0.0f;
  m0[i] = mv;
#pragma unroll
  for (int c = 0; c < 4; ++c) {
    size_t idx = ((size_t)b * 4 + c) * S + s;
    x0[idx] = feat[idx] * mv;
  }
}

// ---------------------------------------------------------------------------
// Mask dilation by the 4^3 kernel footprint (stride-2 down or transpose-up).
// ---------------------------------------------------------------------------
__global__ void k_dilate_mask(const float* __restrict__ min_,
                              float* __restrict__ mout,
                              int Din, int Dout, int trans) {
  int Sout = Dout * Dout * Dout;
  int i = blockIdx.x * 256 + threadIdx.x;   // over B*Sout
  if (i >= 2 * Sout) return;
  int b = i / Sout, v = i - b * Sout;
  int ow = v % Dout; int t = v / Dout;
  int oh = t % Dout; int od = t / Dout;
  const float* __restrict__ mb = min_ + (size_t)b * Din * Din * Din;
  int DD = Din * Din;
  float any = 0.0f;
  for (int kd = 0; kd < 4; ++kd)
    for (int kh = 0; kh < 4; ++kh)
      for (int kw = 0; kw < 4; ++kw) {
        int id, ih, iw; bool ok;
        if (!trans) {
          id = 2 * od - 1 + kd; ih = 2 * oh - 1 + kh; iw = 2 * ow - 1 + kw;
          ok = true;
        } else {
          ok = ((od + kd) & 1) && ((oh + kh) & 1) && ((ow + kw) & 1);
          id = (od + 1 - kd) >> 1; ih = (oh + 1 - kh) >> 1; iw = (ow + 1 - kw) >> 1;
        }
        if (ok && (unsigned)id < (unsigned)Din && (unsigned)ih < (unsigned)Din &&
            (unsigned)iw < (unsigned)Din)
          if (mb[id * DD + ih * Din + iw] > 0.0f) any = 1.0f;
      }
  mout[i] = any;
}

// ---------------------------------------------------------------------------
// Implicit-GEMM conv via V_WMMA_F32_16X16X4_F32, 32x32 macro-tile per wave
// (2x2 WMMA tiles, 4 f32 accumulators). K = Cin*64, ordered
// k = ((ci*4+kd)*4+kh)*4+kw so consecutive K-steps walk the weight row.
//
// Per-lane spatial tap offsets (invariant across ci) are precomputed into
// LDS: offs[vtile][lane][ (kd*4+kh)*2 + u ], -1 = invalid (zero-filled tap).
// Lane's B-operand kw values are {2*half, 2*half+1} per the f32 WMMA VGPR
// layout (lanes<16 hold K0/K1, lanes>=16 hold K2/K3); N = lane&15.
//
// TRANS=false: down conv k4 s2 p1 (id = 2*od-1+kd).
// TRANS=true : transpose conv k4 s2 p1 (tap valid iff (od+kd) odd,
//              id = (od+1-kd)/2) zero-padded into the full 4^3 tap space.
// Epilogue: y = LeakyReLU((acc + bias[co]) * m2[vox]).
// ---------------------------------------------------------------------------
template <bool TRANS>
__global__ __launch_bounds__(32) void k_conv_wmma(
    const float* __restrict__ x, const float* __restrict__ w,
    const float* __restrict__ bias, const float* __restrict__ m2,
    float* __restrict__ y, int Cin, int Cout, int Din, int Dout) {
  __shared__ int offs[2][32][32];   // 8 KB

  const int lane = threadIdx.x;
  const int half = lane >> 4;       // 0: K{0,1}, 1: K{2,3}
  const int l16  = lane & 15;
  const int b    = blockIdx.z;
  const int Sout = Dout * Dout * Dout;
  const int DD   = Din * Din;
  const int Sin  = DD * Din;

  // The two N (voxel) tiles of this wave's 32x32 macro-tile.
  int vox[2], od[2], oh[2], ow[2];
#pragma unroll
  for (int v = 0; v < 2; ++v) {
    vox[v] = blockIdx.y * 32 + v * 16 + l16;
    int t = vox[v];
    ow[v] = t % Dout; t /= Dout;
    oh[v] = t % Dout;
    od[v] = t / Dout;
  }

  // Precompute this lane's 32 spatial tap offsets per N-tile.
#pragma unroll
  for (int v = 0; v < 2; ++v) {
    for (int j = 0; j < 32; ++j) {
      const int kw = half * 2 + (j & 1);
      const int kd = j >> 3;          // j = ((kd*4+kh)*2 + u)
      const int kh = (j >> 1) & 3;
      int off = -1;
      if (!TRANS) {
        const int id = 2 * od[v] - 1 + kd;
        const int ih = 2 * oh[v] - 1 + kh;
        const int iw = 2 * ow[v] - 1 + kw;
        if ((unsigned)id < (unsigned)Din && (unsigned)ih < (unsigned)Din &&
            (unsigned)iw < (unsigned)Din)
          off = id * DD + ih * Din + iw;
      } else {
        if (((od[v] + kd) & 1) && ((oh[v] + kh) & 1) && ((ow[v] + kw) & 1)) {
          const int id = (od[v] + 1 - kd) >> 1;
          const int ih = (oh[v] + 1 - kh) >> 1;
          const int iw = (ow[v] + 1 - kw) >> 1;
          if ((unsigned)id < (unsigned)Din && (unsigned)ih < (unsigned)Din &&
              (unsigned)iw < (unsigned)Din)
            off = id * DD + ih * Din + iw;
        }
      }
      offs[v][lane][j] = off;
    }
  }

  const float* __restrict__ xb  = x + (size_t)b * Cin * Sin;
  const float* __restrict__ wr0 = w + (size_t)(blockIdx.x * 32 + l16) * (Cin * 64);
  const float* __restrict__ wr1 = wr0 + (size_t)16 * (Cin * 64);
  const int2* __restrict__ t0 = (const int2*)&offs[0][lane][0];
  const int2* __restrict__ t1 = (const int2*)&offs[1][lane][0];

  v8f acc00 = {0.f, 0.f, 0.f, 0.f, 0.f, 0.f, 0.f, 0.f};
  v8f acc01 = acc00, acc10 = acc00, acc11 = acc00;

  for (int ci = 0; ci < Cin; ++ci) {
    const float* __restrict__ xc = xb + (size_t)ci * Sin;
    const float* __restrict__ w0 = wr0 + ci * 64;
    const float* __restrict__ w1 = wr1 + ci * 64;
    __builtin_prefetch(w0 + 64, 0, 1);   // next ci's weight row
    __builtin_prefetch(w1 + 64, 0, 1);
#pragma unroll 4
    for (int kk = 0; kk < 16; ++kk) {    // (kd,kh) pairs: K-step of 4
      const int kl = kk * 4 + half * 2;
      const v2f a0 = *(const v2f*)(w0 + kl);
      const v2f a1 = *(const v2f*)(w1 + kl);
      const int2 o0 = t0[kk];
      const int2 o1 = t1[kk];
      v2f b0, b1;
      b0.x = (o0.x >= 0) ? xc[o0.x] : 0.0f;
      b0.y = (o0.y >= 0) ? xc[o0.y] : 0.0f;
      b1.x = (o1.x >= 0) ? xc[o1.x] : 0.0f;
      b1.y = (o1.y >= 0) ? xc[o1.y] : 0.0f;
      acc00 = __builtin_amdgcn_wmma_f32_16x16x4_f32(false, a0, false, b0,
                                                    (short)0, acc00, false, false);
      acc01 = __builtin_amdgcn_wmma_f32_16x16x4_f32(false, a0, false, b1,
                                                    (short)0, acc01, false, false);
      acc10 = __builtin_amdgcn_wmma_f32_16x16x4_f32(false, a1, false, b0,
                                                    (short)0, acc10, false, false);
      acc11 = __builtin_amdgcn_wmma_f32_16x16x4_f32(false, a1, false, b1,
                                                    (short)0, acc11, false, false);
    }
  }

  // C/D layout: VGPR r -> M = r + 8*half, N = l16.
  const float mv0 = m2[(size_t)b * Sout + vox[0]];
  const float mv1 = m2[(size_t)b * Sout + vox[1]];
#pragma unroll
  for (int r = 0; r < 8; ++r) {
    const int co0 = blockIdx.x * 32 + r + half * 8;
    const int co1 = co0 + 16;
    const float bi0 = bias[co0];
    const float bi1 = bias[co1];
    float v;
    v = (acc00[r] + bi0) * mv0; v = (v >= 0.f) ? v : SLOPE * v;
    y[((size_t)b * Cout + co0) * Sout + vox[0]] = v;
    v = (acc01[r] + bi0) * mv1; v = (v >= 0.f) ? v : SLOPE * v;
    y[((size_t)b * Cout + co0) * Sout + vox[1]] = v;
    v = (acc10[r] + bi1) * mv0; v = (v >= 0.f) ? v : SLOPE * v;
    y[((size_t)b * Cout + co1) * Sout + vox[0]] = v;
    v = (acc11[r] + bi1) * mv1; v = (v >= 0.f) ? v : SLOPE * v;
    y[((size_t)b * Cout + co1) * Sout + vox[1]] = v;
  }
}

// ---------------------------------------------------------------------------
// BatchNorm-over-active-voxels statistics. stats[0:128)=sum, [128:256)=sumsq,
// [256]=active count. y is exactly zero off-mask so plain sums suffice.
// ---------------------------------------------------------------------------
__global__ void k_zero_stats(float* __restrict__ stats) {
  if (threadIdx.x < 512) stats[threadIdx.x] = 0.0f;
}

__global__ void k_mask_count(const float* __restrict__ m,
                             float* __restrict__ stats, int n) {
  __shared__ float red[256];
  int t = threadIdx.x;
  int i = blockIdx.x * 256 + t;
  red[t] = (i < n) ? m[i] : 0.0f;
  __syncthreads();
  for (int off = 128; off > 0; off >>= 1) {
    if (t < off) red[t] += red[t + off];
    __syncthreads();
  }
  if (t == 0) atomicAdd(&stats[256], red[0]);
}

__global__ void k_stats(const float* __restrict__ y, float* __restrict__ stats,
                        int C, int S) {
  __shared__ float s1[256];
  __shared__ float s2[256];
  int t = threadIdx.x;
  size_t base = ((size_t)blockIdx.z * C + blockIdx.y) * S;
  float v = y[base + (size_t)blockIdx.x * 256 + t];   // S is a multiple of 256
  s1[t] = v;
  s2[t] = v * v;
  __syncthreads();
  for (int off = 128; off > 0; off >>= 1) {
    if (t < off) { s1[t] += s1[t + off]; s2[t] += s2[t + off]; }
    __syncthreads();
  }
  if (t == 0) {
    atomicAdd(&stats[blockIdx.y], s1[0]);
    atomicAdd(&stats[128 + blockIdx.y], s2[0]);
  }
}

__global__ void k_bn_apply(const float* __restrict__ y,
                           const float* __restrict__ m,
                           const float* __restrict__ stats,
                           const float* __restrict__ g,
                           const float* __restrict__ be,
                           float* __restrict__ xout, int C, int S) {
  size_t total = (size_t)2 * C * S;
  size_t i = (size_t)blockIdx.x * 256 + threadIdx.x;
  if (i >= total) return;
  int s = (int)(i % S);
  int c = (int)((i / S) % C);
  int b = (int)(i / ((size_t)S * C));
  float cnt  = fmaxf(stats[256], 1.0f);
  float mean = stats[c] / cnt;
  float var  = stats[128 + c] / cnt - mean * mean;
  float rstd = rsqrtf(var + BN_EPS);
  float mv   = m[(size_t)b * S + s];
  xout[i] = ((y[i] - mean) * rstd * g[c] + be[c]) * mv;
}

// ---------------------------------------------------------------------------
// Final dense 1x1x1 conv as WMMA GEMM: out[b][po][v] = fw[po][:]·x[b][:][v]+fb
// ---------------------------------------------------------------------------
__global__ __launch_bounds__(32) void k_gemm1x1_wmma(
    const float* __restrict__ x, const float* __restrict__ w,
    const float* __restrict__ bias, float* __restrict__ out,
    int Cin, int Cout, int S) {
  const int lane = threadIdx.x;
  const int half = lane >> 4;
  const int l16  = lane & 15;
  const int b    = blockIdx.z;
  const int vox  = blockIdx.y * 16 + l16;
  const float* __restrict__ xb   = x + (size_t)b * Cin * S;
  const float* __restrict__ wrow = w + (size_t)(blockIdx.x * 16 + l16) * Cin;

  v8f acc = {0.f, 0.f, 0.f, 0.f, 0.f, 0.f, 0.f, 0.f};
  for (int k0 = 0; k0 < Cin; k0 += 4) {
    const int ka = k0 + half * 2;
    const v2f a = *(const v2f*)(wrow + ka);
    v2f bb;
    bb.x = xb[(size_t)ka * S + vox];
    bb.y = xb[(size_t)(ka + 1) * S + vox];
    acc = __builtin_amdgcn_wmma_f32_16x16x4_f32(false, a, false, bb,
                                                (short)0, acc, false, false);
  }
#pragma unroll
  for (int r = 0; r < 8; ++r) {
    const int co = blockIdx.x * 16 + r + half * 8;
    out[((size_t)b * Cout + co) * S + vox] = acc[r] + bias[co];
  }
}

// ---------------------------------------------------------------------------
// Host-side block driver (all launches on `stream`, no allocation).
// ---------------------------------------------------------------------------
static void run_block(const float* xin, const float* min_, float* mout,
                      float* ybuf, float* xout, const float* w, const float* bb,
                      const float* g, const float* be, int Cin, int Cout,
                      int Din, int Dout, bool trans, float* stats,
                      hipStream_t stream) {
  const int Sout = Dout * Dout * Dout;
  k_dilate_mask<<<(2 * Sout + 255) / 256, 256, 0, stream>>>(min_, mout, Din,
                                                            Dout, trans ? 1 : 0);
  dim3 cg(Cout / 32, Sout / 32, 2);   // 32x32 macro-tiles
  if (trans)
    k_conv_wmma<true><<<cg, 32, 0, stream>>>(xin, w, bb, mout, ybuf, Cin, Cout,
                                             Din, Dout);
  else
    k_conv_wmma<false><<<cg, 32, 0, stream>>>(xin, w, bb, mout, ybuf, Cin, Cout,
                                              Din, Dout);
  k_zero_stats<<<1, 512, 0, stream>>>(stats);
  k_mask_count<<<(2 * Sout + 255) / 256, 256, 0, stream>>>(mout, stats,
                                                           2 * Sout);
  k_stats<<<dim3(Sout / 256, Cout, 2), 256, 0, stream>>>(ybuf, stats, Cout,
                                                         Sout);
  size_t total = (size_t)2 * Cout * Sout;
  k_bn_apply<<<(unsigned)((total + 255) / 256), 256, 0, stream>>>(
      ybuf, mout, stats, g, be, xout, Cout, Sout);
}

extern "C" void kernel_launch(void* const* d_in, const int* in_sizes, int n_in,
                              void* d_out, int out_size, void* d_ws,
                              size_t ws_size, hipStream_t stream) {
  (void)in_sizes; (void)n_in; (void)out_size; (void)ws_size;
  const float* feat = (const float*)d_in[0];
  const float* mask = (const float*)d_in[1];
  const float* w1  = (const float*)d_in[2];  const float* b1  = (const float*)d_in[3];
  const float* g1  = (const float*)d_in[4];  const float* be1 = (const float*)d_in[5];
  const float* w2  = (const float*)d_in[6];  const float* b2  = (const float*)d_in[7];
  const float* g2  = (const float*)d_in[8];  const float* be2 = (const float*)d_in[9];
  const float* w3  = (const float*)d_in[10]; const float* b3  = (const float*)d_in[11];
  const float* g3  = (const float*)d_in[12]; const float* be3 = (const float*)d_in[13];
  const float* tw1 = (const float*)d_in[14]; const float* tb1 = (const float*)d_in[15];
  const float* tg1 = (const float*)d_in[16]; const float* tbe1= (const float*)d_in[17];
  const float* tw2 = (const float*)d_in[18]; const float* tb2 = (const float*)d_in[19];
  const float* tg2 = (const float*)d_in[20]; const float* tbe2= (const float*)d_in[21];
  const float* fw  = (const float*)d_in[22]; const float* fb  = (const float*)d_in[23];
  float* out = (float*)d_out;

  // Workspace carve (floats): ~52 MiB total.
  float* p  = (float*)d_ws;
  float* XA = p; p += 4194304;   // ping
  float* XB = p; p += 4194304;   // pong (holds final 2*64*32^3)
  float* Y  = p; p += 4194304;   // pre-BN activations
  float* MA = p; p += 524288;    // mask ping (holds 2*64^3)
  float* MB = p; p += 524288;    // mask pong
  float* ST = p; p += 512;       // BN stats: sum[128], sumsq[128], cnt

  const int S64 = 64 * 64 * 64;
  k_binarize<<<(2 * S64 + 255) / 256, 256, 0, stream>>>(feat, mask, XA, MA, S64);

  // Encoder: 64^3 -> 32^3 -> 16^3 -> 8^3
  run_block(XA, MA, MB, Y, XB, w1, b1, g1, be1,   4,  32, 64, 32, false, ST, stream);
  run_block(XB, MB, MA, Y, XA, w2, b2, g2, be2,  32,  64, 32, 16, false, ST, stream);
  run_block(XA, MA, MB, Y, XB, w3, b3, g3, be3,  64, 128, 16,  8, false, ST, stream);
  // Decoder: 8^3 -> 16^3 -> 32^3
  run_block(XB, MB, MA, Y, XA, tw1, tb1, tg1, tbe1, 128, 128,  8, 16, true, ST, stream);
  run_block(XA, MA, MB, Y, XB, tw2, tb2, tg2, tbe2, 128,  64, 16, 32, true, ST, stream);

  // Final dense 1x1x1 conv: 64 -> 32 at 32^3
  const int S32 = 32 * 32 * 32;
  k_gemm1x1_wmma<<<dim3(32 / 16, S32 / 16, 2), 32, 0, stream>>>(XB, fw, fb, out,
                                                                64, 32, S32);
}